// GritMessagePassingLayer_55937654063569
// MI455X (gfx1250) — compile-verified
//
#include <hip/hip_runtime.h>
#include <math.h>

#define NN   50000
#define EE   600000
#define HIDD 128
#define NH   8
#define DDIM 16
#define LN_EPS 1e-5f

typedef float v2f __attribute__((ext_vector_type(2)));
typedef float v8f __attribute__((ext_vector_type(8)));

// ---------- CDNA5 async global->LDS copy (GLOBAL_LOAD_ASYNC_TO_LDS_B128, ASYNCcnt)
__device__ __forceinline__ void async_ld_b128(const float* gsrc, const float* lds_dst) {
  unsigned int lo = (unsigned int)(unsigned long long)lds_dst; // LDS byte addr = flat[31:0]
  asm volatile("global_load_async_to_lds_b128 %0, %1, off"
               :: "v"(lo), "v"(gsrc) : "memory");
}
__device__ __forceinline__ void wait_asynccnt0() {
#if __has_builtin(__builtin_amdgcn_s_wait_asynccnt)
  __builtin_amdgcn_s_wait_asynccnt(0);
#else
  asm volatile("s_wait_asynccnt 0" ::: "memory");
#endif
}
// copy nfloats (multiple of 1024) contiguous floats into LDS, 16B per lane per step
__device__ __forceinline__ void async_tile_copy(float* lds, const float* g, int nfloats) {
  for (int i = threadIdx.x * 4; i < nfloats; i += 256 * 4)
    async_ld_b128(g + i, lds + i);
}

// Accumulate a 16x16 f32 tile: As = LDS tile (16 x K, row stride lda),
// B = global row-major (K x ldb), output columns [n0, n0+16).
// Uses V_WMMA_F32_16X16X4_F32 over the K dimension.
__device__ __forceinline__ v8f wmma_acc_16x16(const float* __restrict__ As, int lda,
                                              const float* __restrict__ B, int ldb,
                                              int n0, int K, v8f acc) {
  const int lane = threadIdx.x & 31;
  const int half = lane >> 4;   // 0: K pair {k,k+1}; 1: K pair {k+2,k+3}
  const int mn   = lane & 15;   // M for A, N for B
  for (int k = 0; k < K; k += 4) {
    const int kk = k + half * 2;
    v2f a, b;
    a.x = As[mn * lda + kk];
    a.y = As[mn * lda + kk + 1];
    b.x = B[(size_t)kk * ldb + n0 + mn];
    b.y = B[(size_t)(kk + 1) * ldb + n0 + mn];
    acc = __builtin_amdgcn_wmma_f32_16x16x4_f32(false, a, false, b, (short)0, acc, false, false);
  }
  return acc;
}

// ---------------- K0: zero the atomic accumulators (agg, rowV, ssum contiguous)
__global__ __launch_bounds__(256) void k_zero(float* p, size_t n) {
  size_t i = (size_t)blockIdx.x * 256 + threadIdx.x;
  if (i < n) p[i] = 0.f;
}

// ---------------- K1: Qh/Kh/Vh = x @ {WQ,WK,WV}
__global__ __launch_bounds__(256) void k_qkv(const float* __restrict__ x,
                                             const float* __restrict__ WQ,
                                             const float* __restrict__ WK,
                                             const float* __restrict__ WV,
                                             float* __restrict__ Qh,
                                             float* __restrict__ Kh,
                                             float* __restrict__ Vh) {
  __shared__ alignas(16) float As[16 * HIDD];
  const int r0 = blockIdx.x * 16;
  async_tile_copy(As, x + (size_t)r0 * HIDD, 16 * HIDD);
  wait_asynccnt0();
  __syncthreads();
  const int w = threadIdx.x >> 5, lane = threadIdx.x & 31;
  const int half = lane >> 4, mn = lane & 15;
  const int n0 = w * 16;
  v8f aq = {}, ak = {}, av = {};
  aq = wmma_acc_16x16(As, HIDD, WQ, HIDD, n0, HIDD, aq);
  ak = wmma_acc_16x16(As, HIDD, WK, HIDD, n0, HIDD, ak);
  av = wmma_acc_16x16(As, HIDD, WV, HIDD, n0, HIDD, av);
  for (int i = 0; i < 8; i++) {
    const size_t o = (size_t)(r0 + i + 8 * half) * HIDD + n0 + mn;
    Qh[o] = aq[i]; Kh[o] = ak[i]; Vh[o] = av[i];
  }
}

// ---------------- K2: fused edge kernel
// Eh = conn@WE (128->256), conn_e = relu(signed_sqrt((Qh[dst]+Kh[src])*Ew)+Eb),
// score = clip(conn_e . Aw), p = exp(score), ssum[dst] += p
__global__ __launch_bounds__(256) void k_edge(const float* __restrict__ conn,
                                              const int* __restrict__ ei,
                                              const float* __restrict__ Qh,
                                              const float* __restrict__ Kh,
                                              const float* __restrict__ WE,
                                              const float* __restrict__ Aw,
                                              float* __restrict__ conn_e,
                                              float* __restrict__ pbuf,
                                              float* __restrict__ ssum) {
  __shared__ alignas(16) float Cs[16 * HIDD];   // conn tile (GEMM A)
  __shared__ alignas(16) float Ms[16 * HIDD];   // msg1, then conn_e
  __shared__ alignas(16) float Ehs[16 * 256];   // Eh tile (Ew | Eb)
  __shared__ int dsts[16], srcs[16];
  const int e0 = blockIdx.x * 16;
  // async DMA of the conn tile into LDS; overlap with index + gather loads
  async_tile_copy(Cs, conn + (size_t)e0 * HIDD, 16 * HIDD);
  if (threadIdx.x < 16) {
    dsts[threadIdx.x] = ei[e0 + threadIdx.x];
    srcs[threadIdx.x] = ei[EE + e0 + threadIdx.x];
  }
  __syncthreads();            // dsts/srcs visible
  for (int i = threadIdx.x; i < 16 * HIDD; i += 256) {
    const int e = i >> 7, c = i & 127;
    Ms[i] = Qh[(size_t)dsts[e] * HIDD + c] + Kh[(size_t)srcs[e] * HIDD + c];
  }
  wait_asynccnt0();
  __syncthreads();            // Cs + Ms complete
  const int w = threadIdx.x >> 5, lane = threadIdx.x & 31;
  const int half = lane >> 4, mn = lane & 15;
  v8f a0 = {}, a1 = {};
  a0 = wmma_acc_16x16(Cs, HIDD, WE, 256, w * 32, HIDD, a0);
  a1 = wmma_acc_16x16(Cs, HIDD, WE, 256, w * 32 + 16, HIDD, a1);
  for (int i = 0; i < 8; i++) {
    const int r = i + 8 * half;
    Ehs[r * 256 + w * 32 + mn]      = a0[i];
    Ehs[r * 256 + w * 32 + 16 + mn] = a1[i];
  }
  __syncthreads();
  for (int i = threadIdx.x; i < 16 * HIDD; i += 256) {
    const int e = i >> 7, c = i & 127;
    const float ew = Ehs[e * 256 + c], eb = Ehs[e * 256 + 128 + c];
    const float m = Ms[i] * ew;
    const float ss = (m > 0.f) ? sqrtf(m) : ((m < 0.f) ? -sqrtf(-m) : 0.f);
    float v = ss + eb;
    v = v > 0.f ? v : 0.f;
    Ms[i] = v;
    conn_e[(size_t)e0 * HIDD + i] = v;
  }
  __syncthreads();
  if (threadIdx.x < 128) {
    const int e = threadIdx.x >> 3, h = threadIdx.x & 7;
    float s = 0.f;
    for (int d = 0; d < DDIM; d++) s += Ms[e * HIDD + h * DDIM + d] * Aw[d * NH + h];
    s = fminf(fmaxf(s, -5.f), 5.f);
    const float p = expf(s);
    pbuf[(size_t)(e0 + e) * NH + h] = p;
    atomicAdd(&ssum[(size_t)dsts[e] * NH + h], p);
  }
}

// ---------------- K3: scatter-accumulate agg and rowV
__global__ __launch_bounds__(256) void k_scatter(const int* __restrict__ ei,
                                                 const float* __restrict__ Vh,
                                                 const float* __restrict__ conn_e,
                                                 const float* __restrict__ pbuf,
                                                 const float* __restrict__ ssum,
                                                 float* __restrict__ agg,
                                                 float* __restrict__ rowV) {
  const size_t gid = (size_t)blockIdx.x * 256 + threadIdx.x;
  if (gid >= (size_t)EE * HIDD) return;
  const int e = (int)(gid >> 7);
  const int c = (int)(gid & 127);
  const int h = c >> 4;
  const int dst = ei[e], src = ei[EE + e];
  const float wgt = pbuf[(size_t)e * NH + h] / (ssum[(size_t)dst * NH + h] + 1e-16f);
  atomicAdd(&agg[(size_t)dst * HIDD + c], Vh[(size_t)src * HIDD + c] * wgt);
  atomicAdd(&rowV[(size_t)dst * HIDD + c], conn_e[gid] * wgt);
}

// ---------------- K4: node epilogue (Bw mix, deg, Ho GEMM, LN1, FFN, LN2)
__global__ __launch_bounds__(256) void k_node_out(const float* __restrict__ x,
                                                  const float* __restrict__ Qh,
                                                  const float* __restrict__ agg,
                                                  const float* __restrict__ rowV,
                                                  const float* __restrict__ Bw,
                                                  const float* __restrict__ log_deg,
                                                  const float* __restrict__ deg_coef,
                                                  const float* __restrict__ Ho_w,
                                                  const float* __restrict__ Ho_b,
                                                  const float* __restrict__ g1,
                                                  const float* __restrict__ b1n,
                                                  const float* __restrict__ W1,
                                                  const float* __restrict__ bb1,
                                                  const float* __restrict__ W2,
                                                  const float* __restrict__ bb2,
                                                  const float* __restrict__ g2,
                                                  const float* __restrict__ b2n,
                                                  float* __restrict__ hout) {
  __shared__ alignas(16) float A[16 * HIDD];   // GEMM A operand / LN1 output (h_in2)
  __shared__ alignas(16) float H1[16 * HIDD];  // intermediates
  __shared__ alignas(16) float T[16 * 256];    // FFN hidden
  __shared__ float mu[16], rs[16];
  const int r0 = blockIdx.x * 16;
  const int tid = threadIdx.x;
  for (int i = tid; i < 16 * HIDD; i += 256) {
    const int r = i >> 7, c = i & 127, h = c >> 4, d = c & 15;
    const int n = r0 + r;
    float rv = 0.f;
    for (int dd = 0; dd < DDIM; dd++)
      rv += rowV[(size_t)n * HIDD + h * DDIM + dd] * Bw[dd * (NH * DDIM) + h * DDIM + d];
    const float ha = Qh[(size_t)n * HIDD + c] + agg[(size_t)n * HIDD + c] + rv;
    const float ld = log_deg[n];
    A[i] = ha * (deg_coef[c * 2] + ld * deg_coef[c * 2 + 1]);
  }
  __syncthreads();
  const int w = tid >> 5, lane = tid & 31, half = lane >> 4, mn = lane & 15;
  v8f acc = {};
  acc = wmma_acc_16x16(A, HIDD, Ho_w, HIDD, w * 16, HIDD, acc);
  for (int i = 0; i < 8; i++) {
    const int r = i + 8 * half, c = w * 16 + mn;
    H1[r * HIDD + c] = acc[i] + Ho_b[c] + x[(size_t)(r0 + r) * HIDD + c];
  }
  __syncthreads();
  if (tid < 16) {
    float m = 0.f;
    for (int c = 0; c < HIDD; c++) m += H1[tid * HIDD + c];
    m /= HIDD;
    float v = 0.f;
    for (int c = 0; c < HIDD; c++) { float d = H1[tid * HIDD + c] - m; v += d * d; }
    mu[tid] = m; rs[tid] = rsqrtf(v / HIDD + LN_EPS);
  }
  __syncthreads();
  for (int i = tid; i < 16 * HIDD; i += 256) {
    const int r = i >> 7, c = i & 127;
    A[i] = (H1[i] - mu[r]) * rs[r] * g1[c] + b1n[c];   // h_in2
  }
  __syncthreads();
  v8f f0 = {}, f1 = {};
  f0 = wmma_acc_16x16(A, HIDD, W1, 256, w * 32, HIDD, f0);
  f1 = wmma_acc_16x16(A, HIDD, W1, 256, w * 32 + 16, HIDD, f1);
  for (int i = 0; i < 8; i++) {
    const int r = i + 8 * half;
    const int c0 = w * 32 + mn, c1 = w * 32 + 16 + mn;
    float t0 = f0[i] + bb1[c0], t1 = f1[i] + bb1[c1];
    T[r * 256 + c0] = t0 > 0.f ? t0 : 0.f;
    T[r * 256 + c1] = t1 > 0.f ? t1 : 0.f;
  }
  __syncthreads();
  v8f g = {};
  g = wmma_acc_16x16(T, 256, W2, HIDD, w * 16, 256, g);
  for (int i = 0; i < 8; i++) {
    const int r = i + 8 * half, c = w * 16 + mn;
    H1[r * HIDD + c] = A[r * HIDD + c] + g[i] + bb2[c];
  }
  __syncthreads();
  if (tid < 16) {
    float m = 0.f;
    for (int c = 0; c < HIDD; c++) m += H1[tid * HIDD + c];
    m /= HIDD;
    float v = 0.f;
    for (int c = 0; c < HIDD; c++) { float d = H1[tid * HIDD + c] - m; v += d * d; }
    mu[tid] = m; rs[tid] = rsqrtf(v / HIDD + LN_EPS);
  }
  __syncthreads();
  for (int i = tid; i < 16 * HIDD; i += 256) {
    const int r = i >> 7, c = i & 127;
    hout[(size_t)r0 * HIDD + i] = (H1[i] - mu[r]) * rs[r] * g2[c] + b2n[c];
  }
}

// ---------------- K5: edge epilogue (Eo GEMM, residual, LN1e)
__global__ __launch_bounds__(256) void k_edge_out(const float* __restrict__ conn,
                                                   const float* __restrict__ conn_e,
                                                   const float* __restrict__ Eo_w,
                                                   const float* __restrict__ Eo_b,
                                                   const float* __restrict__ ge,
                                                   const float* __restrict__ be,
                                                   float* __restrict__ eout) {
  __shared__ alignas(16) float A[16 * HIDD];
  __shared__ alignas(16) float O[16 * HIDD];
  __shared__ float mu[16], rs[16];
  const int e0 = blockIdx.x * 16;
  const int tid = threadIdx.x;
  async_tile_copy(A, conn_e + (size_t)e0 * HIDD, 16 * HIDD);
  wait_asynccnt0();
  __syncthreads();
  const int w = tid >> 5, lane = tid & 31, half = lane >> 4, mn = lane & 15;
  v8f acc = {};
  acc = wmma_acc_16x16(A, HIDD, Eo_w, HIDD, w * 16, HIDD, acc);
  for (int i = 0; i < 8; i++) {
    const int r = i + 8 * half, c = w * 16 + mn;
    O[r * HIDD + c] = acc[i] + Eo_b[c] + conn[(size_t)(e0 + r) * HIDD + c];
  }
  __syncthreads();
  if (tid < 16) {
    float m = 0.f;
    for (int c = 0; c < HIDD; c++) m += O[tid * HIDD + c];
    m /= HIDD;
    float v = 0.f;
    for (int c = 0; c < HIDD; c++) { float d = O[tid * HIDD + c] - m; v += d * d; }
    mu[tid] = m; rs[tid] = rsqrtf(v / HIDD + LN_EPS);
  }
  __syncthreads();
  for (int i = tid; i < 16 * HIDD; i += 256) {
    const int r = i >> 7, c = i & 127;
    eout[(size_t)e0 * HIDD + i] = (O[i] - mu[r]) * rs[r] * ge[c] + be[c];
  }
}

extern "C" void kernel_launch(void* const* d_in, const int* in_sizes, int n_in,
                              void* d_out, int out_size, void* d_ws, size_t ws_size,
                              hipStream_t stream) {
  const float* x        = (const float*)d_in[0];
  const float* conn     = (const float*)d_in[1];
  const float* log_deg  = (const float*)d_in[2];
  const int*   ei       = (const int*)d_in[3];
  const float* WQ       = (const float*)d_in[4];
  const float* WK       = (const float*)d_in[5];
  const float* WV       = (const float*)d_in[6];
  const float* WE       = (const float*)d_in[7];
  const float* Aw       = (const float*)d_in[8];
  const float* Bw       = (const float*)d_in[9];
  const float* Ho_w     = (const float*)d_in[10];
  const float* Ho_b     = (const float*)d_in[11];
  const float* Eo_w     = (const float*)d_in[12];
  const float* Eo_b     = (const float*)d_in[13];
  const float* deg_coef = (const float*)d_in[14];
  const float* ln1h_g   = (const float*)d_in[15];
  const float* ln1h_b   = (const float*)d_in[16];
  const float* ln1e_g   = (const float*)d_in[17];
  const float* ln1e_b   = (const float*)d_in[18];
  const float* ln2h_g   = (const float*)d_in[19];
  const float* ln2h_b   = (const float*)d_in[20];
  const float* W1       = (const float*)d_in[21];
  const float* b1       = (const float*)d_in[22];
  const float* W2       = (const float*)d_in[23];
  const float* b2       = (const float*)d_in[24];

  float* ws = (float*)d_ws;
  const size_t nh = (size_t)NN * HIDD;
  float* Qh     = ws;                     // N*128
  float* Kh     = Qh + nh;                // N*128
  float* Vh     = Kh + nh;                // N*128
  float* agg    = Vh + nh;                // N*128 (atomic)
  float* rowV   = agg + nh;               // N*128 (atomic)
  float* ssum   = rowV + nh;              // N*8   (atomic)
  float* pbuf   = ssum + (size_t)NN * NH; // E*8
  float* conn_e = pbuf + (size_t)EE * NH; // E*128

  float* hout = (float*)d_out;            // N*128
  float* eout = hout + nh;                // E*128

  // K0: zero atomic accumulators (agg, rowV, ssum are contiguous)
  const size_t nzero = 2 * nh + (size_t)NN * NH;
  k_zero<<<(unsigned)((nzero + 255) / 256), 256, 0, stream>>>(agg, nzero);

  // K1: QKV projections
  k_qkv<<<NN / 16, 256, 0, stream>>>(x, WQ, WK, WV, Qh, Kh, Vh);

  // K2: fused edge attention scores
  k_edge<<<EE / 16, 256, 0, stream>>>(conn, ei, Qh, Kh, WE, Aw, conn_e, pbuf, ssum);

  // K3: softmax-weighted scatter into agg / rowV
  k_scatter<<<(unsigned)(((size_t)EE * HIDD) / 256), 256, 0, stream>>>(ei, Vh, conn_e, pbuf, ssum, agg, rowV);

  // K4: node-side epilogue -> hout
  k_node_out<<<NN / 16, 256, 0, stream>>>(x, Qh, agg, rowV, Bw, log_deg, deg_coef,
                                          Ho_w, Ho_b, ln1h_g, ln1h_b,
                                          W1, b1, W2, b2, ln2h_g, ln2h_b, hout);

  // K5: edge-side epilogue -> eout
  k_edge_out<<<EE / 16, 256, 0, stream>>>(conn, conn_e, Eo_w, Eo_b, ln1e_g, ln1e_b, eout);

  (void)in_sizes; (void)n_in; (void)out_size; (void)ws_size;
}